// RelationAttentionBatch_82566451299182
// MI455X (gfx1250) — compile-verified
//
#include <hip/hip_runtime.h>
#include <hip/hip_bf16.h>

// Problem constants (match reference)
static constexpr int B = 4;
static constexpr int L = 64;
static constexpr int D = 128;
static constexpr int REL_NUM = 50;

typedef __attribute__((ext_vector_type(2))) float v2f;
typedef __attribute__((ext_vector_type(8))) float v8f;

// ---------------------------------------------------------------------------
// Kernel 1: QT[b,r] = Q[b] (64x128)  @  T[r] (128x128)   -> (64x128)
// One wave (32 threads) per 16x16 output tile, f32 WMMA, K stepped by 4.
// grid.x = B * REL_NUM * (L/16) * (D/16) = 6400 waves.
// ---------------------------------------------------------------------------
__global__ __launch_bounds__(32)
void qt_gemm_kernel(const float* __restrict__ Q,   // (B, L, D)
                    const float* __restrict__ T,   // (REL_NUM, D*D) row-major (d,e)
                    float* __restrict__ QT)        // (B, REL_NUM, L, D)
{
    int id = blockIdx.x;
    const int NT = D / 16;         // 8 n-tiles
    const int MT = L / 16;         // 4 m-tiles
    int nt = id % NT; id /= NT;
    int mt = id % MT; id /= MT;
    int r  = id % REL_NUM; id /= REL_NUM;
    int b  = id;

    const int lane = threadIdx.x;      // 0..31
    const int half = lane >> 4;        // 0: K={0,1}, 1: K={2,3} per WMMA A/B layout
    const int l16  = lane & 15;

    // A fragment source: row M = mt*16 + l16 of Q[b]
    const float* qrow = Q + ((size_t)b * L + (size_t)mt * 16 + l16) * D;
    // B fragment source: column N = nt*16 + l16 of T[r] (row-major, row stride D)
    const float* tcol = T + (size_t)r * D * D + (size_t)nt * 16 + l16;

    v8f acc = {};
#pragma unroll
    for (int k0 = 0; k0 < D; k0 += 4) {
        const int ka = k0 + half * 2;
        // A 16x4 f32: lane holds A[M=l16, ka] and A[M=l16, ka+1] (contiguous -> b64)
        v2f a;
        a[0] = qrow[ka];
        a[1] = qrow[ka + 1];
        // B 4x16 f32: lane holds B[ka, N=l16] and B[ka+1, N=l16] (stride-D)
        v2f bf;
        bf[0] = tcol[(size_t)ka * D];
        bf[1] = tcol[(size_t)(ka + 1) * D];

        if ((k0 & 15) == 0) {
            // speculative prefetch of the T panel ~4 K-steps ahead (gfx1250 global_prefetch)
            __builtin_prefetch(tcol + (size_t)(k0 + 16) * D, 0, 1);
        }

        // D = A*B + C, full f32:  v_wmma_f32_16x16x4_f32
        acc = __builtin_amdgcn_wmma_f32_16x16x4_f32(
            /*neg_a=*/false, a, /*neg_b=*/false, bf,
            /*c_mod=*/(short)0, acc, /*reuse_a=*/false, /*reuse_b=*/false);
    }

    // C/D 16x16 f32 layout: element g of lane -> M = g + (half?8:0), N = l16
    float* out = QT + (((size_t)b * REL_NUM + r) * L + (size_t)mt * 16) * D + (size_t)nt * 16;
#pragma unroll
    for (int g = 0; g < 8; ++g) {
        const int m = g + half * 8;
        out[(size_t)m * D + l16] = acc[g];
    }
}

// ---------------------------------------------------------------------------
// Kernel 2: per-(b,i) row: gathered scores, masked softmax, p@V, W_out proj.
// One 64-thread block (2 waves) per (b,i). L == 64 threads, each owns one j.
// ---------------------------------------------------------------------------
__global__ __launch_bounds__(64)
void attn_row_kernel(const float* __restrict__ Kmat, // (B, L, D)
                     const float* __restrict__ V,    // (B, L, D)
                     const int*   __restrict__ R,    // (B, L, L)
                     const float* __restrict__ QT,   // (B, REL_NUM, L, D)
                     const float* __restrict__ W,    // (D, D)
                     const float* __restrict__ bias, // (D)
                     float* __restrict__ out,        // (B, L, D)
                     float* __restrict__ p_out)      // (B, L, L)
{
    const int bi = blockIdx.x;     // 0 .. B*L-1
    const int b  = bi / L;
    const int i  = bi % L;
    const int j  = threadIdx.x;    // 0 .. 63

    __shared__ float red[L];       // reduction scratch
    __shared__ float pshare[L];    // softmax probabilities
    __shared__ float ctx[D];       // context row

    // --- score[b,i,j] = QT[b, R[b,i,j], i, :] . K[b, j, :] -------------------
    const int r = R[((size_t)b * L + i) * L + j];
    const float* qt = QT + (((size_t)b * REL_NUM + r) * L + i) * D;
    const float* kr = Kmat + ((size_t)b * L + j) * D;

    float s = 0.0f;
#pragma unroll 8
    for (int d = 0; d < D; ++d) s += qt[d] * kr[d];
    if (r <= 0) s = -1e9f;

    // --- stable softmax over j (64 threads) ---------------------------------
    red[j] = s;
    __syncthreads();
#pragma unroll
    for (int off = 32; off > 0; off >>= 1) {
        if (j < off) red[j] = fmaxf(red[j], red[j + off]);
        __syncthreads();
    }
    const float smax = red[0];
    __syncthreads();

    const float e = __expf(s - smax);
    red[j] = e;
    __syncthreads();
#pragma unroll
    for (int off = 32; off > 0; off >>= 1) {
        if (j < off) red[j] += red[j + off];
        __syncthreads();
    }
    const float inv_sum = 1.0f / red[0];
    const float p = e * inv_sum;

    pshare[j] = p;
    p_out[((size_t)b * L + i) * L + j] = p;
    __syncthreads();

    // --- ctx[b,i,d] = sum_j p[j] * V[b,j,d]; thread owns d=j and d=j+64 ------
    float c0 = 0.0f, c1 = 0.0f;
    const float* vb = V + (size_t)b * L * D;
#pragma unroll 4
    for (int jj = 0; jj < L; ++jj) {
        const float pj = pshare[jj];
        const float* vr = vb + (size_t)jj * D;
        c0 += pj * vr[j];
        c1 += pj * vr[j + 64];
    }
    ctx[j]      = c0;
    ctx[j + 64] = c1;
    __syncthreads();

    // --- out[b,i,d] = bias[d] + sum_e W[d,e] * ctx[e] ------------------------
    float o0 = bias[j];
    float o1 = bias[j + 64];
#pragma unroll 4
    for (int e2 = 0; e2 < D; ++e2) {
        const float ce = ctx[e2];
        o0 += W[(size_t)j * D + e2] * ce;
        o1 += W[(size_t)(j + 64) * D + e2] * ce;
    }
    float* orow = out + ((size_t)b * L + i) * D;
    orow[j]      = o0;
    orow[j + 64] = o1;
}

// ---------------------------------------------------------------------------
// Launch
// ---------------------------------------------------------------------------
extern "C" void kernel_launch(void* const* d_in, const int* in_sizes, int n_in,
                              void* d_out, int out_size, void* d_ws, size_t ws_size,
                              hipStream_t stream) {
    const float* query     = (const float*)d_in[0];   // (B,L,D)
    const float* key       = (const float*)d_in[1];   // (B,L,D)
    const float* value     = (const float*)d_in[2];   // (B,L,D)
    const int*   R         = (const int*)  d_in[3];   // (B,L,L)
    const float* rel_table = (const float*)d_in[4];   // (REL_NUM, D*D)
    const float* W_out     = (const float*)d_in[5];   // (D,D)
    const float* b_out     = (const float*)d_in[6];   // (D)

    float* out   = (float*)d_out;              // (B,L,D) = 32768 floats
    float* p_out = out + (size_t)B * L * D;    // (B,L,L) = 16384 floats

    float* QT = (float*)d_ws;                  // (B, REL_NUM, L, D) = 6.25 MB

    // Phase 1: 200 dense 64x128x128 GEMMs via f32 WMMA
    const int grid1 = B * REL_NUM * (L / 16) * (D / 16);   // 6400 waves
    qt_gemm_kernel<<<grid1, 32, 0, stream>>>(query, rel_table, QT);

    // Phase 2: gathered scores + softmax + PV + projection, one block per (b,i)
    attn_row_kernel<<<B * L, 64, 0, stream>>>(key, value, R, QT, W_out, b_out,
                                              out, p_out);
}